// integralDecoder_79207786873046
// MI455X (gfx1250) — compile-verified
//
#include <hip/hip_runtime.h>
#include <math.h>

typedef __attribute__((ext_vector_type(2))) float v2f;
typedef __attribute__((ext_vector_type(8))) float v8f;

#define P_POINTS 1000
#define WAVES_PER_BLOCK 8
#define BLOCK_THREADS (WAVES_PER_BLOCK * 32)
#define LOG2E 1.4426950408889634f

// tr[i] = sum_p cg[p] * 2^((i-1)*log2(x_p) + (N-i-1)*log2(1-x_p))
// Reduction over p done with V_WMMA_F32_16X16X4_F32 (full fp32 WMMA).
// Base-2 logs in LDS so the inner loop hits v_exp_f32 directly (no log2e mul).
__global__ __launch_bounds__(BLOCK_THREADS)
void integralDecoder_wmma_kernel(const float* __restrict__ s_ptr,
                                 float* __restrict__ out, int N)
{
    __shared__ float s_lx[P_POINTS];    // log2(x_p)
    __shared__ float s_l1mx[P_POINTS];  // log2(1-x_p)   (0 at p=999 to avoid -inf)
    __shared__ float s_cg[P_POINTS];    // trapezoid weight * g(x_p)  (0 at p=999)

    const float s     = s_ptr[0];
    const float dx    = 0.999f / 999.0f;           // linspace(0.001, 1.0, 1000) step
    const float twoNs = 2.0f * (float)N * s;

    for (int p = (int)threadIdx.x; p < P_POINTS; p += BLOCK_THREADS) {
        float x    = 0.001f + dx * (float)p;
        float omx  = 1.0f - x;
        bool  last = (p == P_POINTS - 1);          // x == 1.0
        float lx   = __builtin_amdgcn_logf(x);                         // log2(x)
        float l1mx = last ? 0.0f : __builtin_amdgcn_logf(omx);         // log2(1-x)
        float g    = 1.0f - __builtin_amdgcn_exp2f(-twoNs * omx * LOG2E); // 1-exp(-2Ns(1-x))
        float w    = dx * ((p == 0) ? 0.5f : 1.0f);
        s_lx[p]   = lx;
        s_l1mx[p] = l1mx;
        s_cg[p]   = last ? 0.0f : (w * g);         // kill p=999 contribution
    }
    __syncthreads();

    const int lane   = threadIdx.x & 31;
    const int wave   = threadIdx.x >> 5;
    const int tile   = blockIdx.x * WAVES_PER_BLOCK + wave;
    const int base_i = tile * 16 + 1;              // rows i = base_i .. base_i+15
    const int m      = lane & 15;                  // row within tile (A layout)
    const int koff   = (lane >> 4) * 2;            // lanes0-15 -> K0,K1; lanes16-31 -> K2,K3

    const float fi  = (float)(base_i + m);
    const float em1 = fi - 1.0f;                   // (i-1)
    const float eN  = (float)N - fi - 1.0f;        // (N-i-1)

    v8f acc = {0.f, 0.f, 0.f, 0.f, 0.f, 0.f, 0.f, 0.f};

    // 1000 points in K=4 chunks -> 250 WMMAs per wave; branch-free (EXEC all 1s).
    #pragma unroll 2
    for (int p0 = 0; p0 < P_POINTS; p0 += 4) {
        int pa = p0 + koff;
        int pb = pa + 1;
        v2f a, b;
        a.x = __builtin_amdgcn_exp2f(em1 * s_lx[pa] + eN * s_l1mx[pa]);
        a.y = __builtin_amdgcn_exp2f(em1 * s_lx[pb] + eN * s_l1mx[pb]);
        b.x = s_cg[pa];                            // B row k replicated across all 16 cols
        b.y = s_cg[pb];
        acc = __builtin_amdgcn_wmma_f32_16x16x4_f32(
            /*neg_a=*/false, a, /*neg_b=*/false, b,
            /*c_mod=*/(short)0, acc, /*reuse_a=*/false, /*reuse_b=*/false);
    }

    // All 16 columns of C are identical (B was column-replicated).
    // C layout: VGPR r holds row r (lanes 0-15) and row r+8 (lanes 16-31).
    if ((lane & 15) == 0) {
        int mbase = (lane >> 4) * 8;
        #pragma unroll
        for (int r = 0; r < 8; ++r) {
            int i = base_i + mbase + r;
            if (i <= N - 1) out[i] = acc[r];
        }
    }
    if (blockIdx.x == 0 && threadIdx.x == 0) out[0] = 0.0f;
}

extern "C" void kernel_launch(void* const* d_in, const int* in_sizes, int n_in,
                              void* d_out, int out_size, void* d_ws, size_t ws_size,
                              hipStream_t stream) {
    (void)in_sizes; (void)n_in; (void)d_ws; (void)ws_size;
    const float* s_ptr = (const float*)d_in[0];   // selection coefficient, shape [1]
    float* out = (float*)d_out;
    const int N = out_size;                        // 32768 (== reference N)
    const int tiles  = (N - 1 + 15) / 16;          // 16 rows of i per wave
    const int blocks = (tiles + WAVES_PER_BLOCK - 1) / WAVES_PER_BLOCK;
    integralDecoder_wmma_kernel<<<blocks, BLOCK_THREADS, 0, stream>>>(s_ptr, out, N);
}